// Contextual_Transformer2d_50483045597338
// MI455X (gfx1250) — compile-verified
//
#include <hip/hip_runtime.h>

typedef __attribute__((ext_vector_type(2))) float v2f;
typedef __attribute__((ext_vector_type(8))) float v8f;

#define B_ 8
#define C_ 512
#define N_ 4096
#define EPS_ 1e-6f
#define HT_ 0.25f

static __device__ __forceinline__ v8f wmma4(v2f a, v2f b, v8f c) {
    // D = A(16x4,f32) * B(4x16,f32) + C(16x16,f32)
    return __builtin_amdgcn_wmma_f32_16x16x4_f32(false, a, false, b, (short)0, c, false, false);
}

// ---------------- Stage 1: Q = Wq*x, K = Wk*x (per-batch channel-mix GEMM) ----
// Each wave: 32 (c) x 32 (n) output for BOTH Q and K. 8 WMMA per k-step,
// B fragments shared across Q/K, A fragments shared across the two n-tiles.
__global__ __launch_bounds__(128) void k_qkproj(const float* __restrict__ x,
                                                const float* __restrict__ Wq,
                                                const float* __restrict__ Wk,
                                                float* __restrict__ Q,
                                                float* __restrict__ K) {
    const int lane = threadIdx.x & 31;
    const int wave = threadIdx.x >> 5;
    const int m    = lane & 15;         // A row / B col within tile
    const int kp   = (lane >> 4) << 1;  // k sub-offset {0,2}
    const int half = lane >> 4;
    const int c0 = blockIdx.y * 32;
    const int n0 = blockIdx.x * 128 + wave * 32;
    const int b  = blockIdx.z;
    const float* xb  = x  + (size_t)b * C_ * N_;
    const float* wq0 = Wq + (size_t)(c0 + m) * C_;
    const float* wq1 = Wq + (size_t)(c0 + 16 + m) * C_;
    const float* wk0 = Wk + (size_t)(c0 + m) * C_;
    const float* wk1 = Wk + (size_t)(c0 + 16 + m) * C_;
    v8f q00 = {}, q01 = {}, q10 = {}, q11 = {};
    v8f k00 = {}, k01 = {}, k10 = {}, k11 = {};
    for (int k = 0; k < C_; k += 4) {
        const int kk = k + kp;
        v2f aq0 = *(const v2f*)(wq0 + kk);
        v2f aq1 = *(const v2f*)(wq1 + kk);
        v2f ak0 = *(const v2f*)(wk0 + kk);
        v2f ak1 = *(const v2f*)(wk1 + kk);
        const float* xr0 = xb + (size_t)kk * N_;
        const float* xr1 = xb + (size_t)(kk + 1) * N_;
        v2f b0, b1;
        b0.x = xr0[n0 + m];      b0.y = xr1[n0 + m];
        b1.x = xr0[n0 + 16 + m]; b1.y = xr1[n0 + 16 + m];
        q00 = wmma4(aq0, b0, q00); q01 = wmma4(aq0, b1, q01);
        q10 = wmma4(aq1, b0, q10); q11 = wmma4(aq1, b1, q11);
        k00 = wmma4(ak0, b0, k00); k01 = wmma4(ak0, b1, k01);
        k10 = wmma4(ak1, b0, k10); k11 = wmma4(ak1, b1, k11);
    }
    float* Qb = Q + (size_t)b * C_ * N_;
    float* Kb = K + (size_t)b * C_ * N_;
    #pragma unroll
    for (int j = 0; j < 8; ++j) {
        const int cA = c0 + j + 8 * half;
        const int cB = cA + 16;
        Qb[(size_t)cA * N_ + n0 + m]      = q00[j];
        Qb[(size_t)cA * N_ + n0 + 16 + m] = q01[j];
        Qb[(size_t)cB * N_ + n0 + m]      = q10[j];
        Qb[(size_t)cB * N_ + n0 + 16 + m] = q11[j];
        Kb[(size_t)cA * N_ + n0 + m]      = k00[j];
        Kb[(size_t)cA * N_ + n0 + 16 + m] = k01[j];
        Kb[(size_t)cB * N_ + n0 + m]      = k10[j];
        Kb[(size_t)cB * N_ + n0 + 16 + m] = k11[j];
    }
}

// ---------------- Stage 2: per-row inverse L2 norms of Q and K ----------------
__global__ __launch_bounds__(256) void k_rownorm(const float* __restrict__ Q,
                                                 const float* __restrict__ K,
                                                 float* __restrict__ invQ,
                                                 float* __restrict__ invK) {
    __shared__ float sq[256], sk[256];
    const int row = blockIdx.x;                     // b*C + c
    const float* qr = Q + (size_t)row * N_;
    const float* kr = K + (size_t)row * N_;
    float aq = 0.f, ak = 0.f;
    for (int n = threadIdx.x; n < N_; n += 256) {
        float v = qr[n]; aq += v * v;
        float w = kr[n]; ak += w * w;
    }
    sq[threadIdx.x] = aq; sk[threadIdx.x] = ak;
    __syncthreads();
    for (int s = 128; s > 0; s >>= 1) {
        if (threadIdx.x < s) {
            sq[threadIdx.x] += sq[threadIdx.x + s];
            sk[threadIdx.x] += sk[threadIdx.x + s];
        }
        __syncthreads();
    }
    if (threadIdx.x == 0) { invQ[row] = rsqrtf(sq[0]); invK[row] = rsqrtf(sk[0]); }
}

// ---------------- Stage 3: M[c][d] = <Qhat_c, Khat_d> over N -------------------
// 2x2 register-blocked: 4 contiguous b64 fragment loads -> 4 WMMAs.
__global__ __launch_bounds__(128) void k_qkt(const float* __restrict__ Q,
                                             const float* __restrict__ K,
                                             const float* __restrict__ invQ,
                                             const float* __restrict__ invK,
                                             float* __restrict__ M) {
    const int lane = threadIdx.x & 31;
    const int wave = threadIdx.x >> 5;
    const int m    = lane & 15;
    const int kp   = (lane >> 4) << 1;
    const int half = lane >> 4;
    const int c0 = blockIdx.y * 32;
    const int d0 = blockIdx.x * 128 + wave * 32;
    const int b  = blockIdx.z;
    const size_t bOff = (size_t)b * C_ * N_;
    const float* q0 = Q + bOff + (size_t)(c0 + m) * N_;
    const float* q1 = Q + bOff + (size_t)(c0 + 16 + m) * N_;
    const float* k0 = K + bOff + (size_t)(d0 + m) * N_;
    const float* k1 = K + bOff + (size_t)(d0 + 16 + m) * N_;
    v8f a00 = {}, a01 = {}, a10 = {}, a11 = {};
    for (int k = 0; k < N_; k += 4) {
        v2f aA = *(const v2f*)(q0 + k + kp);
        v2f aB = *(const v2f*)(q1 + k + kp);
        v2f bA = *(const v2f*)(k0 + k + kp);
        v2f bB = *(const v2f*)(k1 + k + kp);
        a00 = wmma4(aA, bA, a00); a01 = wmma4(aA, bB, a01);
        a10 = wmma4(aB, bA, a10); a11 = wmma4(aB, bB, a11);
    }
    float* Mb = M + (size_t)b * C_ * C_;
    const float ik0 = invK[b * C_ + d0 + m];
    const float ik1 = invK[b * C_ + d0 + 16 + m];
    #pragma unroll
    for (int j = 0; j < 8; ++j) {
        const int cA = c0 + j + 8 * half;
        const int cB = cA + 16;
        const float iqA = invQ[b * C_ + cA];
        const float iqB = invQ[b * C_ + cB];
        Mb[(size_t)cA * C_ + d0 + m]      = a00[j] * iqA * ik0;
        Mb[(size_t)cA * C_ + d0 + 16 + m] = a01[j] * iqA * ik1;
        Mb[(size_t)cB * C_ + d0 + m]      = a10[j] * iqB * ik0;
        Mb[(size_t)cB * C_ + d0 + 16 + m] = a11[j] * iqB * ik1;
    }
}

// ---------------- Stage 4: column stats of (1 - M) over c ---------------------
__global__ __launch_bounds__(128) void k_colstat(const float* __restrict__ M,
                                                 float* __restrict__ denom,
                                                 float* __restrict__ shift) {
    __shared__ float smax[128], smin[128];
    const int idx = blockIdx.x;                 // b*C + d
    const int b = idx / C_;
    const int d = idx - b * C_;
    const float* Mb = M + (size_t)b * C_ * C_;
    float mx = -3.4e38f, mn = 3.4e38f;
    for (int c = threadIdx.x; c < C_; c += 128) {
        const float v = 1.f - Mb[(size_t)c * C_ + d];
        mx = fmaxf(mx, v);
        mn = fminf(mn, v);
    }
    smax[threadIdx.x] = mx; smin[threadIdx.x] = mn;
    __syncthreads();
    for (int s = 64; s > 0; s >>= 1) {
        if (threadIdx.x < s) {
            smax[threadIdx.x] = fmaxf(smax[threadIdx.x], smax[threadIdx.x + s]);
            smin[threadIdx.x] = fminf(smin[threadIdx.x], smin[threadIdx.x + s]);
        }
        __syncthreads();
    }
    if (threadIdx.x == 0) {
        const float dn = smax[0] + EPS_;
        denom[idx] = dn;
        // max over c of (1 - ((1-M)/dn)/h) corresponds to the column min of (1-M)
        shift[idx] = 1.f - (smin[0] / dn) * (1.f / HT_);
    }
}

// ---------------- Stage 5: column-wise softmax, in place ----------------------
__global__ __launch_bounds__(128) void k_colsoftmax(float* __restrict__ M,
                                                    const float* __restrict__ denom,
                                                    const float* __restrict__ shift) {
    __shared__ float ssum[128];
    const int idx = blockIdx.x;                 // b*C + d
    const int b = idx / C_;
    const int d = idx - b * C_;
    float* Mb = M + (size_t)b * C_ * C_;
    const float inv_dn = 1.f / denom[idx];
    const float sh = shift[idx];
    float z = 0.f;
    for (int c = threadIdx.x; c < C_; c += 128) {
        const float m4 = 1.f - ((1.f - Mb[(size_t)c * C_ + d]) * inv_dn) * (1.f / HT_);
        z += expf(m4 - sh);
    }
    ssum[threadIdx.x] = z;
    __syncthreads();
    for (int s = 64; s > 0; s >>= 1) {
        if (threadIdx.x < s) ssum[threadIdx.x] += ssum[threadIdx.x + s];
        __syncthreads();
    }
    const float invz = 1.f / ssum[0];
    for (int c = threadIdx.x; c < C_; c += 128) {
        const size_t o = (size_t)c * C_ + d;
        const float m4 = 1.f - ((1.f - Mb[o]) * inv_dn) * (1.f / HT_);
        Mb[o] = expf(m4 - sh) * invz;
    }
}

// ---------------- Stage 6: row-wise L1 normalization, in place ----------------
__global__ __launch_bounds__(256) void k_rowl1(float* __restrict__ M) {
    __shared__ float ssum[256];
    const int row = blockIdx.x;                 // b*C + c (rows are contiguous)
    float* r = M + (size_t)row * C_;
    float s = 0.f;
    for (int d = threadIdx.x; d < C_; d += 256) s += r[d];
    ssum[threadIdx.x] = s;
    __syncthreads();
    for (int t = 128; t > 0; t >>= 1) {
        if (threadIdx.x < t) ssum[threadIdx.x] += ssum[threadIdx.x + t];
        __syncthreads();
    }
    const float inv = 1.f / (ssum[0] + EPS_);
    for (int d = threadIdx.x; d < C_; d += 256) r[d] *= inv;
}

// ---------------- Stage 7: G = M * Wv  (folds V projection into M) ------------
__global__ __launch_bounds__(128) void k_mwv(const float* __restrict__ M,
                                             const float* __restrict__ Wv,
                                             float* __restrict__ G) {
    const int lane = threadIdx.x & 31;
    const int wave = threadIdx.x >> 5;
    const int m    = lane & 15;
    const int kp   = (lane >> 4) << 1;
    const int half = lane >> 4;
    const int c0 = blockIdx.y * 32;
    const int e0 = blockIdx.x * 128 + wave * 32;
    const int b  = blockIdx.z;
    const float* a0 = M + (size_t)b * C_ * C_ + (size_t)(c0 + m) * C_;
    const float* a1 = M + (size_t)b * C_ * C_ + (size_t)(c0 + 16 + m) * C_;
    v8f g00 = {}, g01 = {}, g10 = {}, g11 = {};
    for (int k = 0; k < C_; k += 4) {
        const int kk = k + kp;
        v2f aA = *(const v2f*)(a0 + kk);
        v2f aB = *(const v2f*)(a1 + kk);
        const float* wr0 = Wv + (size_t)kk * C_;
        const float* wr1 = Wv + (size_t)(kk + 1) * C_;
        v2f b0, b1;
        b0.x = wr0[e0 + m];      b0.y = wr1[e0 + m];
        b1.x = wr0[e0 + 16 + m]; b1.y = wr1[e0 + 16 + m];
        g00 = wmma4(aA, b0, g00); g01 = wmma4(aA, b1, g01);
        g10 = wmma4(aB, b0, g10); g11 = wmma4(aB, b1, g11);
    }
    float* Gb = G + (size_t)b * C_ * C_;
    #pragma unroll
    for (int j = 0; j < 8; ++j) {
        const int cA = c0 + j + 8 * half;
        const int cB = cA + 16;
        Gb[(size_t)cA * C_ + e0 + m]      = g00[j];
        Gb[(size_t)cA * C_ + e0 + 16 + m] = g01[j];
        Gb[(size_t)cB * C_ + e0 + m]      = g10[j];
        Gb[(size_t)cB * C_ + e0 + 16 + m] = g11[j];
    }
}

// ---------------- Stage 8: y = x + gamma * (G @ x) ----------------------------
__global__ __launch_bounds__(128) void k_out(const float* __restrict__ x,
                                             const float* __restrict__ G,
                                             const float* __restrict__ gamma,
                                             float* __restrict__ y) {
    const int lane = threadIdx.x & 31;
    const int wave = threadIdx.x >> 5;
    const int m    = lane & 15;
    const int kp   = (lane >> 4) << 1;
    const int half = lane >> 4;
    const int c0 = blockIdx.y * 32;
    const int n0 = blockIdx.x * 128 + wave * 32;
    const int b  = blockIdx.z;
    const float* xb = x + (size_t)b * C_ * N_;
    const float* g0 = G + (size_t)b * C_ * C_ + (size_t)(c0 + m) * C_;
    const float* g1 = G + (size_t)b * C_ * C_ + (size_t)(c0 + 16 + m) * C_;
    v8f o00 = {}, o01 = {}, o10 = {}, o11 = {};
    for (int k = 0; k < C_; k += 4) {
        const int kk = k + kp;
        v2f aA = *(const v2f*)(g0 + kk);
        v2f aB = *(const v2f*)(g1 + kk);
        const float* xr0 = xb + (size_t)kk * N_;
        const float* xr1 = xb + (size_t)(kk + 1) * N_;
        v2f b0, b1;
        b0.x = xr0[n0 + m];      b0.y = xr1[n0 + m];
        b1.x = xr0[n0 + 16 + m]; b1.y = xr1[n0 + 16 + m];
        o00 = wmma4(aA, b0, o00); o01 = wmma4(aA, b1, o01);
        o10 = wmma4(aB, b0, o10); o11 = wmma4(aB, b1, o11);
    }
    const float g = gamma[0];
    float* yb = y + (size_t)b * C_ * N_;
    #pragma unroll
    for (int j = 0; j < 8; ++j) {
        const int cA = c0 + j + 8 * half;
        const int cB = cA + 16;
        const size_t o0 = (size_t)cA * N_ + n0 + m;
        const size_t o1 = (size_t)cA * N_ + n0 + 16 + m;
        const size_t o2 = (size_t)cB * N_ + n0 + m;
        const size_t o3 = (size_t)cB * N_ + n0 + 16 + m;
        yb[o0] = xb[o0] + g * o00[j];
        yb[o1] = xb[o1] + g * o01[j];
        yb[o2] = xb[o2] + g * o10[j];
        yb[o3] = xb[o3] + g * o11[j];
    }
}

extern "C" void kernel_launch(void* const* d_in, const int* in_sizes, int n_in,
                              void* d_out, int out_size, void* d_ws, size_t ws_size,
                              hipStream_t stream) {
    (void)in_sizes; (void)n_in; (void)out_size; (void)ws_size;
    const float* x     = (const float*)d_in[0];
    const float* Wq    = (const float*)d_in[1];
    const float* Wk    = (const float*)d_in[2];
    const float* Wv    = (const float*)d_in[3];
    const float* gamma = (const float*)d_in[4];
    float* y = (float*)d_out;

    char* ws = (char*)d_ws;
    const size_t qkBytes = (size_t)B_ * C_ * N_ * sizeof(float);   // 64 MB each
    const size_t mBytes  = (size_t)B_ * C_ * C_ * sizeof(float);   // 8 MB each
    float* Q     = (float*)(ws);
    float* K     = (float*)(ws + qkBytes);
    float* M     = (float*)(ws + 2 * qkBytes);
    float* G     = (float*)(ws + 2 * qkBytes + mBytes);
    float* invQ  = (float*)(ws + 2 * qkBytes + 2 * mBytes);
    float* invK  = invQ  + B_ * C_;
    float* denom = invK  + B_ * C_;
    float* shift = denom + B_ * C_;

    k_qkproj    <<<dim3(N_ / 128, C_ / 32, B_), 128, 0, stream>>>(x, Wq, Wk, Q, K);
    k_rownorm   <<<B_ * C_, 256, 0, stream>>>(Q, K, invQ, invK);
    k_qkt       <<<dim3(C_ / 128, C_ / 32, B_), 128, 0, stream>>>(Q, K, invQ, invK, M);
    k_colstat   <<<B_ * C_, 128, 0, stream>>>(M, denom, shift);
    k_colsoftmax<<<B_ * C_, 128, 0, stream>>>(M, denom, shift);
    k_rowl1     <<<B_ * C_, 256, 0, stream>>>(M);
    k_mwv       <<<dim3(C_ / 128, C_ / 32, B_), 128, 0, stream>>>(M, Wv, G);
    k_out       <<<dim3(N_ / 128, C_ / 32, B_), 128, 0, stream>>>(x, G, gamma, y);
}